// BahdanauAttention_56899726737844
// MI455X (gfx1250) — compile-verified
//
#include <hip/hip_runtime.h>

#define BB 8
#define TQ 256
#define TK 256
#define DQk 512
#define DVv 512
#define UU 256

typedef __attribute__((ext_vector_type(2))) float v2f;
typedef __attribute__((ext_vector_type(8))) float v8f;

// Hardware tanh on CDNA5 (v_tanh_f32) when the builtin exists; otherwise a
// 2-trans-op formula that saturates correctly at +-1.
__device__ __forceinline__ float fast_tanh(float x) {
#if __has_builtin(__builtin_amdgcn_tanhf)
  return __builtin_amdgcn_tanhf(x);
#else
  float e = __builtin_amdgcn_exp2f(x * 2.885390081777927f);  // exp2(2x*log2e)
  return 1.f - 2.f * __builtin_amdgcn_rcpf(e + 1.f);
#endif
}

// ---------------------------------------------------------------------------
// Kernel 1: projection GEMM  out[N,256] = X[N,512] @ W[512,256] + bias
// Block = 256 threads (8 waves), computes a 16-row output stripe.
// Uses V_WMMA_F32_16X16X4_F32 (f32 in / f32 out, full precision).
// ---------------------------------------------------------------------------
__global__ __launch_bounds__(256)
void proj_gemm_kernel(const float* __restrict__ X, const float* __restrict__ W,
                      const float* __restrict__ bias, float* __restrict__ out) {
  __shared__ float sA[16][DQk + 1];  // padded: stride 513 -> conflict-free
  const int t = threadIdx.x;
  const int row0 = blockIdx.x * 16;

  for (int i = t; i < 16 * DQk; i += 256) {
    int r = i >> 9;          // /512
    int c = i & (DQk - 1);   // %512
    sA[r][c] = X[(row0 + r) * DQk + c];
  }
  __syncthreads();

  const int wave = t >> 5;
  const int lane = t & 31;
  const int lrow = (lane < 16) ? lane : (lane - 16);
  const int kof  = (lane < 16) ? 0 : 2;

  for (int nt = wave; nt < UU / 16; nt += 8) {
    const int n0 = nt * 16;
    v8f acc = {};
    for (int k0 = 0; k0 < DQk; k0 += 4) {
      v2f a, b;
      // A 16x4: lane=M, vgpr0/1 = K {0,1} (lanes0-15) or {2,3} (lanes16-31)
      a.x = sA[lrow][k0 + kof];
      a.y = sA[lrow][k0 + kof + 1];
      // B 4x16: lane=N, vgpr0/1 = K {0,1} / {2,3}
      b.x = W[(k0 + kof) * UU + n0 + lrow];
      b.y = W[(k0 + kof + 1) * UU + n0 + lrow];
      acc = __builtin_amdgcn_wmma_f32_16x16x4_f32(false, a, false, b,
                                                  (short)0, acc, false, false);
    }
    const int nc = n0 + lrow;
    const float bv = bias[nc];
#pragma unroll
    for (int r = 0; r < 8; ++r) {
      // D layout: vgpr r = row r (lanes0-15) / row r+8 (lanes16-31)
      const int M = (lane < 16) ? r : (r + 8);
      out[(row0 + M) * UU + nc] = acc[r] + bv;
    }
  }
}

// ---------------------------------------------------------------------------
// Kernel 2: fused additive-attention core. One block per (b, 16-query tile).
//  logits[q,k] = sum_u V[u] * tanh(sq[q,u] + sv[k,u])   (no 537MB tensor!)
//  softmax over k, write attention weights, then context = attn @ values
//  via f32 WMMA. ~99KB dynamic LDS (<< 320KB/WGP, 3 blocks/WGP).
// ---------------------------------------------------------------------------
#define SVP (UU + 4)   // s_sv row stride: 16B-aligned rows for b128 LDS loads

__global__ __launch_bounds__(256)
void attn_kernel(const float* __restrict__ sq, const float* __restrict__ sv,
                 const float* __restrict__ values, const float* __restrict__ Vw,
                 float* __restrict__ ctx_out, float* __restrict__ attn_out) {
  extern __shared__ float smem[];
  float* s_sq   = smem;                        // 16 * 257   (16448 B, 16B-mult)
  float* s_sv   = s_sq + 16 * (UU + 1);        // 64 * 260   (16B-aligned rows)
  float* s_attn = s_sv + 64 * SVP;             // 16 * 257
  float* s_V    = s_attn + 16 * (TK + 1);      // 256
  float* s_red  = s_V + UU;                    // 16 * 17
  float* s_stat = s_red + 16 * 17;             // 32 (16 max + 16 1/sum)

  const int t  = threadIdx.x;
  const int b  = blockIdx.x >> 4;
  const int q0 = (blockIdx.x & 15) * 16;

  for (int i = t; i < 16 * UU; i += 256) {
    int r = i >> 8, c = i & 255;
    s_sq[r * (UU + 1) + c] = sq[((b * TQ) + q0 + r) * UU + c];
  }
  if (t < UU) s_V[t] = Vw[t];

  for (int ck = 0; ck < TK / 64; ++ck) {
    __syncthreads();
    for (int i = t; i < 64 * UU; i += 256) {
      int r = i >> 8, c = i & 255;
      s_sv[r * SVP + c] = sv[((b * TK) + ck * 64 + r) * UU + c];
    }
    __syncthreads();
#pragma unroll
    for (int i = 0; i < 4; ++i) {
      const int p  = i * 256 + t;
      const int q  = p >> 6;    // 0..15 (wave-uniform -> LDS broadcast)
      const int kl = p & 63;    // 0..63 (per-lane distinct)
      const float* rq = s_sq + q * (UU + 1);
      const float4* rv4 = reinterpret_cast<const float4*>(s_sv + kl * SVP);
      float acc = 0.f;
#pragma unroll 2
      for (int u4 = 0; u4 < UU / 4; ++u4) {
        const float4 v = rv4[u4];        // ds_load_b128
        const int u = u4 * 4;
        acc = fmaf(fast_tanh(rq[u + 0] + v.x), s_V[u + 0], acc);
        acc = fmaf(fast_tanh(rq[u + 1] + v.y), s_V[u + 1], acc);
        acc = fmaf(fast_tanh(rq[u + 2] + v.z), s_V[u + 2], acc);
        acc = fmaf(fast_tanh(rq[u + 3] + v.w), s_V[u + 3], acc);
      }
      s_attn[q * (TK + 1) + ck * 64 + kl] = acc;
    }
  }
  __syncthreads();

  // ---- row softmax over k (16 threads per row); V_b cancels in softmax ----
  const int r = t >> 4, c16 = t & 15;
  float m = -3.0e38f;
  for (int j = 0; j < 16; ++j)
    m = fmaxf(m, s_attn[r * (TK + 1) + c16 + j * 16]);
  s_red[r * 17 + c16] = m;
  __syncthreads();
  if (c16 == 0) {
    float mm = s_red[r * 17];
    for (int j = 1; j < 16; ++j) mm = fmaxf(mm, s_red[r * 17 + j]);
    s_stat[r] = mm;
  }
  __syncthreads();
  const float rowmax = s_stat[r];
  const float LOG2E = 1.4426950408889634f;
  float ssum = 0.f;
  for (int j = 0; j < 16; ++j) {
    int col = c16 + j * 16;
    float e = __builtin_amdgcn_exp2f((s_attn[r * (TK + 1) + col] - rowmax) * LOG2E);
    s_attn[r * (TK + 1) + col] = e;
    ssum += e;
  }
  s_red[r * 17 + c16] = ssum;
  __syncthreads();
  if (c16 == 0) {
    float s = 0.f;
    for (int j = 0; j < 16; ++j) s += s_red[r * 17 + j];
    s_stat[16 + r] = __builtin_amdgcn_rcpf(s);
  }
  __syncthreads();
  const float rs = s_stat[16 + r];
  for (int j = 0; j < 16; ++j) {
    int col = c16 + j * 16;
    float w = s_attn[r * (TK + 1) + col] * rs;
    s_attn[r * (TK + 1) + col] = w;
    attn_out[((b * TQ) + q0 + r) * TK + col] = w;
  }
  __syncthreads();

  // ---- context: attn[16x256] @ values[b][256x512] via f32 WMMA ----
  const int wave = t >> 5;
  const int lane = t & 31;
  const int lrow = (lane < 16) ? lane : (lane - 16);
  const int kof  = (lane < 16) ? 0 : 2;
  const float* vb = values + (size_t)b * TK * DVv;

  for (int nt = wave; nt < DVv / 16; nt += 8) {
    const int n0 = nt * 16;
    v8f acc = {};
    for (int k0 = 0; k0 < TK; k0 += 4) {
      v2f a, bmat;
      a.x = s_attn[lrow * (TK + 1) + k0 + kof];
      a.y = s_attn[lrow * (TK + 1) + k0 + kof + 1];
      bmat.x = vb[(k0 + kof) * DVv + n0 + lrow];
      bmat.y = vb[(k0 + kof + 1) * DVv + n0 + lrow];
      acc = __builtin_amdgcn_wmma_f32_16x16x4_f32(false, a, false, bmat,
                                                  (short)0, acc, false, false);
    }
#pragma unroll
    for (int rr = 0; rr < 8; ++rr) {
      const int M = (lane < 16) ? rr : (rr + 8);
      ctx_out[((b * TQ) + q0 + M) * DVv + n0 + lrow] = acc[rr];
    }
  }
}

// ---------------------------------------------------------------------------
extern "C" void kernel_launch(void* const* d_in, const int* in_sizes, int n_in,
                              void* d_out, int out_size, void* d_ws, size_t ws_size,
                              hipStream_t stream) {
  const float* query = (const float*)d_in[0];   // [B,TQ,DQ]
  const float* values = (const float*)d_in[1];  // [B,TK,DV]
  const float* W1_w = (const float*)d_in[2];    // [DV,U]
  const float* W1_b = (const float*)d_in[3];    // [U]
  const float* W2_w = (const float*)d_in[4];    // [DQ,U]
  const float* W2_b = (const float*)d_in[5];    // [U]
  const float* V_w  = (const float*)d_in[6];    // [U,1]
  // d_in[7] = V_b: constant logit shift, cancels in softmax.

  float* out  = (float*)d_out;
  float* ctx  = out;                              // [B,TQ,DV]
  float* attn = out + (size_t)BB * TQ * DVv;      // [B,TQ,TK]

  float* sq = (float*)d_ws;                       // [B*TQ, U]  (2 MB)
  float* sv = sq + (size_t)BB * TQ * UU;          // [B*TK, U]  (2 MB)

  proj_gemm_kernel<<<BB * TQ / 16, 256, 0, stream>>>(query, W2_w, W2_b, sq);
  proj_gemm_kernel<<<BB * TK / 16, 256, 0, stream>>>(values, W1_w, W1_b, sv);

  const size_t smem = (16 * (UU + 1) + 64 * SVP + 16 * (TK + 1) +
                       UU + 16 * 17 + 32) * sizeof(float);
  attn_kernel<<<BB * (TQ / 16), 256, smem, stream>>>(sq, sv, values, V_w, ctx, attn);
}